// ATACloss_6811818132296
// MI455X (gfx1250) — compile-verified
//
#include <hip/hip_runtime.h>
#include <math.h>

typedef __attribute__((ext_vector_type(2))) float v2f;
typedef __attribute__((ext_vector_type(8))) float v8f;

// log(n!) for n = 0..10  (true_counts = round(10*uniform) -> integers 0..10)
__device__ __constant__ float g_logfact[11] = {
    0.0f, 0.0f, 0.6931471806f, 1.7917594692f, 3.1780538303f,
    4.7874917428f, 6.5792512120f, 8.5251613611f, 10.6046029027f,
    12.8018274801f, 15.1044125731f
};

// lgamma for x > 0: shift up then Stirling series. Cold path only.
__device__ __noinline__ float lgamma_pos(float x) {
    float acc = 0.f;
    while (x < 8.f) { acc -= logf(x); x += 1.f; }
    float inv  = 1.f / x;
    float inv2 = inv * inv;
    float ser  = inv * (0.0833333333f - inv2 * (0.00277777778f - inv2 * 0.000793650794f));
    return acc + (x - 0.5f) * logf(x) - x + 0.918938533205f + ser;
}

// ---------------------------------------------------------------------------
// Kernel 1: one wave per 16-row tile. Streams v & logits in the WMMA f32
// A-matrix layout (16x4 per chunk: lanes 0-15 hold K0,K1; lanes 16-31 K2,K3
// of the same rows) and accumulates all row-sum reductions with
// V_WMMA_F32_16X16X4_F32 against an all-ones B matrix (D[m,n] = sum_k A[m,k]).
// Pass 1: sum_v, sum(x*v), sum(lgamma(v+1)), row max.
// Pass 2: sum exp(x - max)  (logits re-read hits L2: whole set < 192 MB).
// ---------------------------------------------------------------------------
__global__ void __launch_bounds__(128)
atac_rowstats_kernel(const float* __restrict__ v, const float* __restrict__ x,
                     float* __restrict__ ws, int B, int L) {
    __shared__ float tbl[11];
    if (threadIdx.x < 11) tbl[threadIdx.x] = g_logfact[threadIdx.x];
    __syncthreads();

    const int lane = threadIdx.x & 31;
    const int wid  = threadIdx.x >> 5;
    const int r0   = (blockIdx.x * 4 + wid) * 16;   // 4 waves / block, 16 rows / wave
    int row = r0 + (lane & 15);
    if (row >= B) row = B - 1;                      // clamp loads; writes are guarded
    const int coff = (lane >> 4) * 2;               // lanes 0-15: K0,K1; lanes 16-31: K2,K3

    const float* vrow = v + (long long)row * L;
    const float* xrow = x + (long long)row * L;

    v8f cv = {}, cxv = {}, clg = {};
    v2f ones; ones.x = 1.f; ones.y = 1.f;
    float mr = -INFINITY;

    const int nch = L >> 2;
    // -------- pass 1: sum_v, sum(x*v), sum lgamma(v+1), running row max ----
#pragma unroll 2
    for (int k = 0; k < nch; ++k) {
        const int c = (k << 2) + coff;
        // scalar-uniform guard: plain s_cmp/s_cbranch, no EXEC manipulation
        if (((k & 7) == 0) && (((k << 2) + 68) < L)) {
            __builtin_prefetch(xrow + c + 64, 0, 1);
            __builtin_prefetch(vrow + c + 64, 0, 1);
        }
        v2f va = *(const v2f*)(vrow + c);
        v2f xa = *(const v2f*)(xrow + c);
        mr = fmaxf(mr, fmaxf(xa.x, xa.y));
        v2f axv; axv.x = xa.x * va.x; axv.y = xa.y * va.y;

        // Branchless lgamma(v+1): unconditional clamped LDS lookup (11 entries
        // -> distinct banks, same-address broadcast => conflict-free), with a
        // single combined execz-skipped fallback for out-of-table values.
        int   i0  = (int)(va.x + 0.5f);
        int   i1  = (int)(va.y + 0.5f);
        int   q0  = min(max(i0, 0), 10);
        int   q1  = min(max(i1, 0), 10);
        float lg0 = tbl[q0];
        float lg1 = tbl[q1];
        bool bad0 = (i0 != q0) || (fabsf(va.x - (float)i0) >= 1e-3f);
        bool bad1 = (i1 != q1) || (fabsf(va.y - (float)i1) >= 1e-3f);
        if (__builtin_expect(bad0 || bad1, 0)) {
            if (bad0) lg0 = lgamma_pos(va.x + 1.f);
            if (bad1) lg1 = lgamma_pos(va.y + 1.f);
        }
        v2f alg; alg.x = lg0; alg.y = lg1;

        cv  = __builtin_amdgcn_wmma_f32_16x16x4_f32(false, va,  false, ones, (short)0, cv,  false, false);
        cxv = __builtin_amdgcn_wmma_f32_16x16x4_f32(false, axv, false, ones, (short)0, cxv, false, false);
        clg = __builtin_amdgcn_wmma_f32_16x16x4_f32(false, alg, false, ones, (short)0, clg, false, false);
    }
    if (L & 3) {                                    // masked tail chunk (dead for L=1000)
        const int cb = L & ~3;
        v2f va = {}, xa = {};
        const bool ok0 = (cb + coff) < L, ok1 = (cb + coff + 1) < L;
        if (ok0) { va.x = vrow[cb + coff];     xa.x = xrow[cb + coff];     mr = fmaxf(mr, xa.x); }
        if (ok1) { va.y = vrow[cb + coff + 1]; xa.y = xrow[cb + coff + 1]; mr = fmaxf(mr, xa.y); }
        v2f axv; axv.x = xa.x * va.x; axv.y = xa.y * va.y;
        v2f alg = {};
        if (ok0) { int i0=(int)(va.x+0.5f); int q0=min(max(i0,0),10);
                   alg.x = ((i0==q0)&&(fabsf(va.x-(float)i0)<1e-3f)) ? tbl[q0] : lgamma_pos(va.x+1.f); }
        if (ok1) { int i1=(int)(va.y+0.5f); int q1=min(max(i1,0),10);
                   alg.y = ((i1==q1)&&(fabsf(va.y-(float)i1)<1e-3f)) ? tbl[q1] : lgamma_pos(va.y+1.f); }
        cv  = __builtin_amdgcn_wmma_f32_16x16x4_f32(false, va,  false, ones, (short)0, cv,  false, false);
        cxv = __builtin_amdgcn_wmma_f32_16x16x4_f32(false, axv, false, ones, (short)0, cxv, false, false);
        clg = __builtin_amdgcn_wmma_f32_16x16x4_f32(false, alg, false, ones, (short)0, clg, false, false);
    }

    // Row max: row (lane&15) lives only in lanes {l, l+16} -> one xor-16 shuffle.
    mr = fmaxf(mr, __shfl_xor(mr, 16, 32));

    // -------- pass 2: sum exp(x - max) (logits re-read from L2) ------------
    v8f cse = {};
#pragma unroll 4
    for (int k = 0; k < nch; ++k) {
        const int c = (k << 2) + coff;
        v2f xa = *(const v2f*)(xrow + c);
        v2f ae; ae.x = __expf(xa.x - mr); ae.y = __expf(xa.y - mr);
        cse = __builtin_amdgcn_wmma_f32_16x16x4_f32(false, ae, false, ones, (short)0, cse, false, false);
    }
    if (L & 3) {
        const int cb = L & ~3;
        v2f ae = {};
        if ((cb + coff)     < L) ae.x = __expf(xrow[cb + coff]     - mr);
        if ((cb + coff + 1) < L) ae.y = __expf(xrow[cb + coff + 1] - mr);
        cse = __builtin_amdgcn_wmma_f32_16x16x4_f32(false, ae, false, ones, (short)0, cse, false, false);
    }

    // -------- write per-row stats --------
    float* ws_sumv  = ws;
    float* ws_sumxv = ws + (size_t)B;
    float* ws_sumlg = ws + 2 * (size_t)B;
    float* ws_sumse = ws + 3 * (size_t)B;
    float* ws_max   = ws + 4 * (size_t)B;

    if (lane < 16) {
        int r = r0 + lane;
        if (r < B) ws_max[r] = mr;
    }
    // C layout: VGPR j of lanes 0-15 = row j; of lanes 16-31 = row j+8
    // (row sum broadcast across all 16 columns, so lane 0 / lane 16 suffice).
    if ((lane & 15) == 0) {
        const int rb = r0 + ((lane >> 4) << 3);
#pragma unroll
        for (int j = 0; j < 8; ++j) {
            int r = rb + j;
            if (r < B) {
                ws_sumv[r]  = cv[j];
                ws_sumxv[r] = cxv[j];
                ws_sumlg[r] = clg[j];
                ws_sumse[r] = cse[j];
            }
        }
    }
}

// ---------------------------------------------------------------------------
// Kernel 2: per-row finish math + deterministic fixed-order tree reduction.
// out = ( sum_r (sum_v - tot_pred)^2  -  sum_r log_prob ) / B
// log_prob = lgamma(sum_v+1) - sum_lg + sum_xv - (max + log(sumexp)) * sum_v
// ---------------------------------------------------------------------------
__global__ void __launch_bounds__(256)
atac_finish_kernel(const float* __restrict__ tot_pred, const float* __restrict__ ws,
                   float* __restrict__ out, int B) {
    __shared__ float red[256];
    const float* ws_sumv  = ws;
    const float* ws_sumxv = ws + (size_t)B;
    const float* ws_sumlg = ws + 2 * (size_t)B;
    const float* ws_sumse = ws + 3 * (size_t)B;
    const float* ws_max   = ws + 4 * (size_t)B;

    const int t = threadIdx.x;
    float acc = 0.f;
    for (int r = t; r < B; r += 256) {
        float sv      = ws_sumv[r];
        float logZ    = ws_max[r] + logf(ws_sumse[r]);
        float logprob = lgamma_pos(sv + 1.f) - ws_sumlg[r] + (ws_sumxv[r] - logZ * sv);
        float d       = sv - tot_pred[r];
        acc += d * d - logprob;
    }
    red[t] = acc;
    __syncthreads();
    for (int s = 128; s > 0; s >>= 1) {
        if (t < s) red[t] += red[t + s];
        __syncthreads();
    }
    if (t == 0) out[0] = red[0] / (float)B;
}

extern "C" void kernel_launch(void* const* d_in, const int* in_sizes, int n_in,
                              void* d_out, int out_size, void* d_ws, size_t ws_size,
                              hipStream_t stream) {
    const float* v  = (const float*)d_in[0];   // true_counts (B,L)
    const float* x  = (const float*)d_in[1];   // logits      (B,L)
    const float* tp = (const float*)d_in[2];   // tot_pred    (B,1)
    const int B = in_sizes[2];
    const int L = in_sizes[0] / B;

    float* ws = (float*)d_ws;                  // needs 5*B floats (320 KB for B=16384)
    const int blocks = (B + 63) / 64;          // 4 waves/block * 16 rows/wave
    atac_rowstats_kernel<<<blocks, 128, 0, stream>>>(v, x, ws, B, L);
    atac_finish_kernel<<<1, 256, 0, stream>>>(tp, ws, (float*)d_out, B);
}